// MultiTaskNAM_52115133170272
// MI455X (gfx1250) — compile-verified
//
#include <hip/hip_runtime.h>
#include <hip/hip_bf16.h>

// Shapes from reference
#define BB 4096
#define FF 64
#define SS 8
#define UU 64
#define H1H 64
#define H2H 32
#define TT 4

typedef __attribute__((ext_vector_type(2))) float v2f;
typedef __attribute__((ext_vector_type(8))) float v8f;

// One workgroup = one (f,s) group x 64 rows of B. 4 waves, 16 rows each.
// Chain: h1 = relu(x*W1+b1); h2 = relu(h1@W2+b2); h3 = relu(h2@W3+b3);
//        sub = h3.W4 + b4  -> workspace (B,F,S)
__global__ __launch_bounds__(128) void nam_subnets_kernel(
    const float* __restrict__ x,
    const float* __restrict__ W1, const float* __restrict__ b1,
    const float* __restrict__ W2, const float* __restrict__ b2,
    const float* __restrict__ W3, const float* __restrict__ b3,
    const float* __restrict__ W4, const float* __restrict__ b4,
    float* __restrict__ sub)
{
    // LDS: weights in WMMA-fragment-friendly (transposed) order + per-wave
    // relayout stages for D->A between GEMMs.
    __shared__ __align__(16) float sW2t[UU * H1H];        // [(u/4)*64 + h]*4 + (u&3)
    __shared__ __align__(16) float sW3t[H1H * H2H];       // [(u/4)*32 + h]*4 + (u&3)
    __shared__ __align__(16) float sW1[UU];
    __shared__ __align__(16) float sB1[UU];
    __shared__ __align__(16) float sB2[H1H];
    __shared__ __align__(16) float sB3[H2H];
    __shared__ __align__(16) float sW4[H2H];
    __shared__ __align__(16) float sH2[4][16 * 68];       // stride 68: conflict-free
    __shared__ __align__(16) float sH3[4][16 * 36];

    const int tid  = threadIdx.x;
    const int lane = tid & 31;
    const int wave = tid >> 5;
    const int wg   = blockIdx.x;
    const int fs   = wg & (FF * SS - 1);   // 512 groups
    const int bt   = wg >> 9;              // 64 B-tiles of 64 rows
    const int f    = fs >> 3;
    const int s    = fs & 7;

    // ---- cooperative weight staging (pre-transposed into fragment order) ----
    const float* W2g = W2 + (size_t)fs * UU * H1H;
    for (int i = tid; i < UU * H1H; i += 128) {
        int u = i >> 6, h = i & 63;
        sW2t[((u >> 2) * H1H + h) * 4 + (u & 3)] = W2g[i];
    }
    const float* W3g = W3 + (size_t)fs * H1H * H2H;
    for (int i = tid; i < H1H * H2H; i += 128) {
        int u = i >> 5, h = i & 31;
        sW3t[((u >> 2) * H2H + h) * 4 + (u & 3)] = W3g[i];
    }
    if (tid < 64) {
        sW1[tid] = W1[(size_t)fs * UU + tid];
        sB1[tid] = b1[(size_t)fs * UU + tid];
        sB2[tid] = b2[(size_t)fs * H1H + tid];
    } else {
        int j = tid - 64;
        if (j < 32) sB3[j] = b3[(size_t)fs * H2H + j];
        else        sW4[j - 32] = W4[(size_t)fs * H2H + (j - 32)];
    }
    const float bias4 = b4[fs];
    __syncthreads();

    const int m    = lane & 15;            // M row within tile / N col for B,C,D
    const int half = lane >> 4;            // K-split of fragment layouts
    const int b0   = bt * 64 + wave * 16;  // first B row of this wave

    // ---- Layer 1: compute h1 directly in A-fragment layout ----
    const float xv = x[(size_t)(b0 + m) * FF + f];
    v2f h1f[16];
#pragma unroll
    for (int c = 0; c < 16; ++c) {
        int k0 = 4 * c + 2 * half;
        h1f[c].x = fmaxf(fmaf(xv, sW1[k0],     sB1[k0]),     0.f);
        h1f[c].y = fmaxf(fmaf(xv, sW1[k0 + 1], sB1[k0 + 1]), 0.f);
    }

    // ---- Layer 2: (16x64) = relu((16x64)@(64x64) + b2) via WMMA f32 ----
    float* myH2 = &sH2[wave][0];
#pragma unroll
    for (int nt = 0; nt < 4; ++nt) {
        v8f acc;
        float cinit = sB2[nt * 16 + m];    // C/D lane indexes N; b2 const in M
#pragma unroll
        for (int v = 0; v < 8; ++v) acc[v] = cinit;
#pragma unroll
        for (int c = 0; c < 16; ++c) {
            v2f bf = *(const v2f*)&sW2t[((c * H1H) + nt * 16 + m) * 4 + 2 * half];
            acc = __builtin_amdgcn_wmma_f32_16x16x4_f32(
                false, h1f[c], false, bf, (short)0, acc, false, false);
        }
#pragma unroll
        for (int v = 0; v < 8; ++v)
            myH2[(v + 8 * half) * 68 + nt * 16 + m] = fmaxf(acc[v], 0.f);
    }

    // D-layout -> A-layout through wave-private LDS (in-order per wave)
    v2f h2f[16];
#pragma unroll
    for (int c = 0; c < 16; ++c)
        h2f[c] = *(const v2f*)&myH2[m * 68 + 4 * c + 2 * half];

    // ---- Layer 3: (16x32) = relu((16x64)@(64x32) + b3) ----
    float* myH3 = &sH3[wave][0];
#pragma unroll
    for (int nt = 0; nt < 2; ++nt) {
        v8f acc;
        float cinit = sB3[nt * 16 + m];
#pragma unroll
        for (int v = 0; v < 8; ++v) acc[v] = cinit;
#pragma unroll
        for (int c = 0; c < 16; ++c) {
            v2f bf = *(const v2f*)&sW3t[((c * H2H) + nt * 16 + m) * 4 + 2 * half];
            acc = __builtin_amdgcn_wmma_f32_16x16x4_f32(
                false, h2f[c], false, bf, (short)0, acc, false, false);
        }
#pragma unroll
        for (int v = 0; v < 8; ++v)
            myH3[(v + 8 * half) * 36 + nt * 16 + m] = fmaxf(acc[v], 0.f);
    }

    // ---- Layer 4: sub[b] = h3[b,:].W4 + b4 (16 lanes, one row each) ----
    if (lane < 16) {
        float a4 = bias4;
#pragma unroll
        for (int n = 0; n < H2H; ++n) a4 = fmaf(myH3[m * 36 + n], sW4[n], a4);
        sub[((size_t)(b0 + m) * FF + f) * SS + s] = a4;
    }
}

// stacked[b,t,f] = sum_s sub[b,f,s]*Wt[f,s,t];  summed[b,t] = sum_f stacked + bias[t]
__global__ __launch_bounds__(64) void nam_mix_kernel(
    const float* __restrict__ sub, const float* __restrict__ Wt,
    const float* __restrict__ bias,
    float* __restrict__ summed, float* __restrict__ stacked)
{
    __shared__ float red[TT][FF];
    const int b = blockIdx.x;
    const int f = threadIdx.x;

    const float* sb = sub + ((size_t)b * FF + f) * SS;
    const float* wt = Wt + (size_t)f * SS * TT;
    float o[TT] = {0.f, 0.f, 0.f, 0.f};
#pragma unroll
    for (int s = 0; s < SS; ++s) {
        float v = sb[s];
#pragma unroll
        for (int t = 0; t < TT; ++t) o[t] = fmaf(v, wt[s * TT + t], o[t]);
    }
#pragma unroll
    for (int t = 0; t < TT; ++t) {
        stacked[((size_t)b * TT + t) * FF + f] = o[t];
        red[t][f] = o[t];
    }
    __syncthreads();
    if (f < TT) {
        float acc = bias[f];
#pragma unroll 8
        for (int i = 0; i < FF; ++i) acc += red[f][i];
        summed[(size_t)b * TT + f] = acc;
    }
}

extern "C" void kernel_launch(void* const* d_in, const int* in_sizes, int n_in,
                              void* d_out, int out_size, void* d_ws, size_t ws_size,
                              hipStream_t stream) {
    (void)in_sizes; (void)n_in; (void)out_size; (void)ws_size;
    const float* x   = (const float*)d_in[0];
    const float* W1  = (const float*)d_in[1];
    const float* b1  = (const float*)d_in[2];
    const float* W2  = (const float*)d_in[3];
    const float* b2  = (const float*)d_in[4];
    const float* W3  = (const float*)d_in[5];
    const float* b3  = (const float*)d_in[6];
    const float* W4  = (const float*)d_in[7];
    const float* b4  = (const float*)d_in[8];
    const float* Wt  = (const float*)d_in[9];
    const float* bia = (const float*)d_in[10];

    float* out     = (float*)d_out;
    float* summed  = out;                     // (B,T) first (return order)
    float* stacked = out + (size_t)BB * TT;   // (B,T,F)
    float* sub     = (float*)d_ws;            // (B,F,S) = 8 MB scratch

    // 512 (f,s) groups x 64 B-tiles of 64 rows; 4 waves/WG, 16 rows/wave
    nam_subnets_kernel<<<(FF * SS) * (BB / 64), 128, 0, stream>>>(
        x, W1, b1, W2, b2, W3, b3, W4, b4, sub);
    nam_mix_kernel<<<BB, FF, 0, stream>>>(sub, Wt, bia, summed, stacked);
}